// UndoNoise_34213709480337
// MI455X (gfx1250) — compile-verified
//
#include <hip/hip_runtime.h>

#define N_NODES 200000
#define N_EDGES 50000
#define N_INC   4000000

typedef __attribute__((ext_vector_type(2))) float v2f;
typedef __attribute__((ext_vector_type(4))) float v4f;
typedef __attribute__((ext_vector_type(8))) float v8f;

// ---------------- degree counts: D[node] += 1, B[edge] += 1 ----------------
__global__ void degrees_kernel(const int* __restrict__ nidx, const int* __restrict__ eidx,
                               float* __restrict__ D, float* __restrict__ B, int n) {
    int i = blockIdx.x * blockDim.x + threadIdx.x;
    if (i >= n) return;
    __builtin_prefetch(&nidx[i + 8192], 0, 1);   // speculative: safe past end
    __builtin_prefetch(&eidx[i + 8192], 0, 1);
    atomicAdd(&D[nidx[i]], 1.0f);
    atomicAdd(&B[eidx[i]], 1.0f);
}

// ---------------- xt = x(N,3) @ W(3,16) via V_WMMA_F32_16X16X4_F32 ---------
// One wave per 16-row tile. K=3 padded to 4 with zeros.
// A 16x4 layout: VGPR0 = {K0 lanes0-15, K2 lanes16-31}, VGPR1 = {K1, K3}.
// B 4x16 mirrors on the K axis; C/D: VGPR r = rows {r, r+8}.
__global__ void gemm_x3_w16(const float* __restrict__ x, const float* __restrict__ W,
                            float* __restrict__ xt, int ntiles) {
    int wave = (blockIdx.x * blockDim.x + threadIdx.x) >> 5;
    int lane = threadIdx.x & 31;
    if (wave >= ntiles) return;          // wave-uniform: EXEC stays all-1s
    int  row0 = wave * 16;
    bool lo   = lane < 16;
    int  m    = lo ? lane : (lane - 16);

    v2f a, b;
    if (lo) { a.x = x[(row0 + m) * 3 + 0]; a.y = x[(row0 + m) * 3 + 1]; }
    else    { a.x = x[(row0 + m) * 3 + 2]; a.y = 0.0f; }          // K=3 pad
    if (lo) { b.x = W[0 * 16 + m];         b.y = W[1 * 16 + m]; }
    else    { b.x = W[2 * 16 + m];         b.y = 0.0f; }          // K=3 pad

    v8f c = {};
    c = __builtin_amdgcn_wmma_f32_16x16x4_f32(false, a, false, b, (short)0, c, false, false);

    int mbase = lo ? 0 : 8;
    #pragma unroll
    for (int r = 0; r < 8; ++r)
        xt[(row0 + mbase + r) * 16 + m] = c[r];
}

// ---------------- xt = h(N,16) @ W(16,3): four chained 16x16x4 WMMAs -------
__global__ void gemm_h16_w3(const float* __restrict__ h, const float* __restrict__ W,
                            float* __restrict__ xt, int ntiles) {
    int wave = (blockIdx.x * blockDim.x + threadIdx.x) >> 5;
    int lane = threadIdx.x & 31;
    if (wave >= ntiles) return;
    int  row0 = wave * 16;
    bool lo   = lane < 16;
    int  m    = lo ? lane : (lane - 16);

    v8f c = {};
    #pragma unroll
    for (int kk = 0; kk < 4; ++kk) {
        int kbase = kk * 4 + (lo ? 0 : 2);
        v2f a, b;
        a.x = h[(row0 + m) * 16 + kbase + 0];
        a.y = h[(row0 + m) * 16 + kbase + 1];
        b.x = (m < 3) ? W[(kbase + 0) * 3 + m] : 0.0f;   // N=3, pad cols 3..15
        b.y = (m < 3) ? W[(kbase + 1) * 3 + m] : 0.0f;
        c = __builtin_amdgcn_wmma_f32_16x16x4_f32(false, a, false, b, (short)0, c, false, false);
    }
    int mbase = lo ? 0 : 8;
    if (m < 3) {
        #pragma unroll
        for (int r = 0; r < 8; ++r)
            xt[(row0 + mbase + r) * 3 + m] = c[r];
    }
}

// ---------------- segment scatter-add over incidence list ------------------
// Rows are F floats; for F%4==0 force float4 gathers (global_load_b128):
// every row base is 16B-aligned by construction.
template <int F>
__global__ void scatter_add(const int* __restrict__ sidx, const int* __restrict__ didx,
                            const float* __restrict__ src, float* __restrict__ dst, int n) {
    int i = blockIdx.x * blockDim.x + threadIdx.x;
    if (i >= n) return;
    __builtin_prefetch(&sidx[i + 8192], 0, 1);
    __builtin_prefetch(&didx[i + 8192], 0, 1);
    int s = sidx[i] * F;
    int d = didx[i] * F;
    if constexpr ((F & 3) == 0) {
        const v4f* s4 = reinterpret_cast<const v4f*>(src + s);
        #pragma unroll
        for (int q = 0; q < F / 4; ++q) {
            v4f v = s4[q];
            atomicAdd(&dst[d + 4 * q + 0], v.x);
            atomicAdd(&dst[d + 4 * q + 1], v.y);
            atomicAdd(&dst[d + 4 * q + 2], v.z);
            atomicAdd(&dst[d + 4 * q + 3], v.w);
        }
    } else {
        #pragma unroll
        for (int f = 0; f < F; ++f)
            atomicAdd(&dst[d + f], src[s + f]);
    }
}

// ---------------- ef[e,:] *= 1/B[e] (0 if empty) ---------------------------
template <int F>
__global__ void edge_scale(const float* __restrict__ B, float* __restrict__ ef, int nedges) {
    int i = blockIdx.x * blockDim.x + threadIdx.x;
    if (i >= nedges) return;
    float bv  = B[i];
    float inv = bv > 0.0f ? 1.0f / bv : 0.0f;
    if constexpr ((F & 3) == 0) {
        v4f* p = reinterpret_cast<v4f*>(ef + i * F);
        #pragma unroll
        for (int q = 0; q < F / 4; ++q) {
            v4f v = p[q];
            v.x *= inv; v.y *= inv; v.z *= inv; v.w *= inv;
            p[q] = v;
        }
    } else {
        #pragma unroll
        for (int f = 0; f < F; ++f)
            ef[i * F + f] *= inv;
    }
}

// ---------------- out[n,:] = out[n,:]/D[n] + bias --------------------------
template <int F>
__global__ void node_finish(const float* __restrict__ D, const float* __restrict__ bias,
                            float* __restrict__ out, int nnodes) {
    int i = blockIdx.x * blockDim.x + threadIdx.x;
    if (i >= nnodes) return;
    float dv  = D[i];
    float inv = dv > 0.0f ? 1.0f / dv : 0.0f;
    if constexpr ((F & 3) == 0) {
        v4f* p = reinterpret_cast<v4f*>(out + i * F);
        #pragma unroll
        for (int q = 0; q < F / 4; ++q) {
            v4f v = p[q];
            v.x = inv * v.x + bias[4 * q + 0];
            v.y = inv * v.y + bias[4 * q + 1];
            v.z = inv * v.z + bias[4 * q + 2];
            v.w = inv * v.w + bias[4 * q + 3];
            p[q] = v;
        }
    } else {
        #pragma unroll
        for (int f = 0; f < F; ++f)
            out[i * F + f] = inv * out[i * F + f] + bias[f];
    }
}

extern "C" void kernel_launch(void* const* d_in, const int* in_sizes, int n_in,
                              void* d_out, int out_size, void* d_ws, size_t ws_size,
                              hipStream_t stream) {
    const float* x   = (const float*)d_in[0];
    const int*   hei = (const int*)  d_in[1];   // [2, N_INC]: row0=node, row1=edge
    const float* W1  = (const float*)d_in[2];   // [3,16]
    const float* b1  = (const float*)d_in[3];   // [16]
    const float* W2  = (const float*)d_in[4];   // [16,3]
    const float* b2  = (const float*)d_in[5];   // [3]
    const int* nidx = hei;
    const int* eidx = hei + N_INC;
    float* out = (float*)d_out;                 // [N_NODES, 3]

    // Workspace carve-up (all offsets 16B-aligned); ~32.8 MB total.
    char*  ws  = (char*)d_ws;
    float* D   = (float*)(ws);                   //  200000 f32 (0.8 MB)
    float* B   = (float*)(ws +   800000);        //   50000 f32 (0.2 MB)
    float* xt1 = (float*)(ws +  1000000);        // 200000*16  (12.8 MB)
    float* ef1 = (float*)(ws + 13800000);        //  50000*16  (3.2 MB)
    float* h   = (float*)(ws + 17000000);        // 200000*16  (12.8 MB)
    float* xt2 = (float*)(ws + 29800000);        // 200000*3   (2.4 MB)
    float* ef2 = (float*)(ws + 32200000);        //  50000*3   (0.6 MB)

    // Re-zero every accumulator each call (graph-capture-safe memsets).
    hipMemsetAsync(D,   0, (N_NODES + N_EDGES) * sizeof(float), stream); // D+B contiguous
    hipMemsetAsync(ef1, 0, N_EDGES * 16 * sizeof(float), stream);
    hipMemsetAsync(h,   0, N_NODES * 16 * sizeof(float), stream);
    hipMemsetAsync(ef2, 0, N_EDGES * 3 * sizeof(float), stream);
    hipMemsetAsync(out, 0, N_NODES * 3 * sizeof(float), stream);

    const int TB = 256;                          // 8 wave32s per block
    const int incBlocks  = (N_INC   + TB - 1) / TB;
    const int edgeBlocks = (N_EDGES + TB - 1) / TB;
    const int nodeBlocks = (N_NODES + TB - 1) / TB;
    const int ntiles     = N_NODES / 16;         // 12500 exactly
    const int gemmBlocks = (ntiles + (TB / 32) - 1) / (TB / 32);

    degrees_kernel<<<incBlocks, TB, 0, stream>>>(nidx, eidx, D, B, N_INC);

    // ---- layer 1: 3 -> 16 ----
    gemm_x3_w16<<<gemmBlocks, TB, 0, stream>>>(x, W1, xt1, ntiles);
    scatter_add<16><<<incBlocks, TB, 0, stream>>>(nidx, eidx, xt1, ef1, N_INC);
    edge_scale<16><<<edgeBlocks, TB, 0, stream>>>(B, ef1, N_EDGES);
    scatter_add<16><<<incBlocks, TB, 0, stream>>>(eidx, nidx, ef1, h, N_INC);
    node_finish<16><<<nodeBlocks, TB, 0, stream>>>(D, b1, h, N_NODES);

    // ---- layer 2: 16 -> 3 ----
    gemm_h16_w3<<<gemmBlocks, TB, 0, stream>>>(h, W2, xt2, ntiles);
    scatter_add<3><<<incBlocks, TB, 0, stream>>>(nidx, eidx, xt2, ef2, N_INC);
    edge_scale<3><<<edgeBlocks, TB, 0, stream>>>(B, ef2, N_EDGES);
    scatter_add<3><<<incBlocks, TB, 0, stream>>>(eidx, nidx, ef2, out, N_INC);
    node_finish<3><<<nodeBlocks, TB, 0, stream>>>(D, b2, out, N_NODES);
}